// OptimalTransport_69801808495371
// MI455X (gfx1250) — compile-verified
//
#include <hip/hip_runtime.h>
#include <math.h>

typedef float v2f __attribute__((ext_vector_type(2)));
typedef float v8f __attribute__((ext_vector_type(8)));

constexpr int S = 2048;
constexpr int Q = 4096;
constexpr int D = 1024;
constexpr float INV_EPS = 20.0f;       // 1/0.05
constexpr float EPSR    = 0.05f;
constexpr float LOG_MU  = -7.6246189861593985f; // -ln(2048)
constexpr float LOG_NU  = -8.3177661667193430f; // -ln(4096)
constexpr float THRESH  = 1e-3f;
constexpr int   MAX_ITER = 50;

// ---- workspace layout (float units) ----
constexpr size_t ZS_OFF  = 0;                       // zs normalized [S*D]
constexpr size_t C_OFF   = ZS_OFF + (size_t)S * D;  // C / pi [S*Q]
constexpr size_t U_OFF   = C_OFF + (size_t)S * Q;   // u  [S]
constexpr size_t UN_OFF  = U_OFF + S;               // u_new [S]
constexpr size_t V_OFF   = UN_OFF + S;              // v  [Q]
constexpr size_t VN_OFF  = V_OFF + Q;               // v_new [Q]
constexpr size_t SQS_OFF = VN_OFF + Q;              // |zs_i|^2 [S]
constexpr size_t SQQ_OFF = SQS_OFF + S;             // |zq_j|^2 [Q]
constexpr size_t RS_OFF  = SQQ_OFF + Q;             // 1/rowsum(pi) [S]
constexpr size_t FLAG_OFF= RS_OFF + S;              // done flag (int)

// ---------- CDNA5 async global->LDS copy (no VGPR staging, ASYNCcnt tracked) ----------
__device__ __forceinline__ void async_copy16(const float* gsrc, float* lds_dst) {
  unsigned lds_off = (unsigned)(uintptr_t)lds_dst;              // addr[31:0] = LDS offset
  unsigned long long gaddr = (unsigned long long)(uintptr_t)gsrc;
  asm volatile("global_load_async_to_lds_b128 %0, %1, off"
               :: "v"(lds_off), "v"(gaddr)
               : "memory");
}
__device__ __forceinline__ void wait_async0() {
  asm volatile("s_wait_asynccnt 0x0" ::: "memory");
}

// ---------- helpers ----------
__device__ __forceinline__ void lse_accum(float& m, float& s, float x) {
  if (x <= m) { s += expf(x - m); }
  else        { s = s * expf(m - x) + 1.0f; m = x; }
}
__device__ __forceinline__ void lse_merge(float& m, float& s, float m2, float s2) {
  float nm = fmaxf(m, m2);
  s = s * expf(m - nm) + s2 * expf(m2 - nm);
  m = nm;
}

// ---------- init ----------
__global__ void ot_init(float* u, float* v, int* done) {
  int t = threadIdx.x;
  for (int i = t; i < S; i += 256) u[i] = 0.0f;
  for (int j = t; j < Q; j += 256) v[j] = 0.0f;
  if (t == 0) *done = 0;
}

// ---------- power transform + L2 normalize, one row per block (float4) ----------
__global__ void ot_normalize(const float* __restrict__ src, float* __restrict__ dst,
                             float* __restrict__ sq) {
  const int row = blockIdx.x;
  const int t = threadIdx.x;
  const size_t base = (size_t)row * D + (size_t)t * 4;
  float4 xv = *(const float4*)&src[base];
  float4 y;
  y.x = sqrtf(xv.x + 1e-6f); y.y = sqrtf(xv.y + 1e-6f);
  y.z = sqrtf(xv.z + 1e-6f); y.w = sqrtf(xv.w + 1e-6f);
  float partial = y.x * y.x + y.y * y.y + y.z * y.z + y.w * y.w;

  __shared__ float red[256];
  red[t] = partial; __syncthreads();
  for (int st = 128; st > 0; st >>= 1) {
    if (t < st) red[t] += red[t + st];
    __syncthreads();
  }
  float sum = red[0];
  float scale = 1.0f / fmaxf(sqrtf(sum), 1e-12f);
  y.x *= scale; y.y *= scale; y.z *= scale; y.w *= scale;
  *(float4*)&dst[base] = y;
  if (t == 0) sq[row] = sum * scale * scale;  // (z*z).sum(1) after normalize
}

// ---------- GEMM1: C = sqs_i + sqq_j - 2 * zs @ zq^T  (fp32 WMMA 16x16x4) ----------
// block tile 64(M) x 128(N), K-chunk 32, async-copy double-buffered LDS
__global__ __launch_bounds__(256)
void ot_gemm_cost(const float* __restrict__ zs, const float* __restrict__ zq,
                  const float* __restrict__ sqs, const float* __restrict__ sqq,
                  float* __restrict__ Cm) {
  __shared__ float As[2][64][36];    // [buf][m][k], 16B-aligned rows, conflict-free
  __shared__ float Bs[2][128][36];   // [buf][n][k]
  const int t    = threadIdx.x;
  const int wave = t >> 5;
  const int lane = t & 31;
  const int half = lane >> 4;        // 0: K 0,1 ; 1: K 2,3
  const int lm   = lane & 15;
  const int mw   = wave & 3;         // m-tile 0..3
  const int nw0  = (wave >> 2) * 4;  // n-tiles nw0..nw0+3

  const int m0 = blockIdx.y * 64;
  const int n0 = blockIdx.x * 128;

  auto issue_chunk = [&](int k0, int buf) {
#pragma unroll
    for (int i = 0; i < 2; ++i) {
      int idx = t + 256 * i, rr = idx >> 3, k4 = idx & 7;
      async_copy16(&zs[(size_t)(m0 + rr) * D + k0 + k4 * 4], &As[buf][rr][k4 * 4]);
    }
#pragma unroll
    for (int i = 0; i < 4; ++i) {
      int idx = t + 256 * i, rr = idx >> 3, k4 = idx & 7;
      async_copy16(&zq[(size_t)(n0 + rr) * D + k0 + k4 * 4], &Bs[buf][rr][k4 * 4]);
    }
  };

  v8f acc[4];
#pragma unroll
  for (int q = 0; q < 4; ++q) acc[q] = v8f{};

  issue_chunk(0, 0);
  wait_async0();
  __syncthreads();

  for (int k0 = 0; k0 < D; k0 += 32) {
    const int buf = (k0 >> 5) & 1;
    const bool more = (k0 + 32) < D;
    if (more) issue_chunk(k0 + 32, buf ^ 1);   // DMA overlaps WMMA below
#pragma unroll
    for (int kk = 0; kk < 32; kk += 4) {
      const int ko = kk + half * 2;
      v2f a; a.x = As[buf][mw * 16 + lm][ko]; a.y = As[buf][mw * 16 + lm][ko + 1];
#pragma unroll
      for (int q = 0; q < 4; ++q) {
        v2f b; b.x = Bs[buf][(nw0 + q) * 16 + lm][ko];
               b.y = Bs[buf][(nw0 + q) * 16 + lm][ko + 1];
        acc[q] = __builtin_amdgcn_wmma_f32_16x16x4_f32(false, a, false, b, (short)0,
                                                       acc[q], false, false);
      }
    }
    if (more) wait_async0();
    __syncthreads();
  }

#pragma unroll
  for (int r = 0; r < 8; ++r) {
    int gi = m0 + mw * 16 + r + half * 8;
    float si = sqs[gi];
#pragma unroll
    for (int q = 0; q < 4; ++q) {
      int j = n0 + (nw0 + q) * 16 + lm;
      Cm[(size_t)gi * Q + j] = si + sqq[j] - 2.0f * acc[q][r];
    }
  }
}

// ---------- Sinkhorn: row logsumexp -> u_new (float4) ----------
__global__ void ot_row(const float* __restrict__ Cm, const float* __restrict__ u,
                       const float* __restrict__ v, float* __restrict__ u_new,
                       const int* __restrict__ done) {
  if (*done) return;
  const int i = blockIdx.x;
  const int t = threadIdx.x;
  const float ui = u[i];
  float m = -INFINITY, s = 0.0f;
  const float4* Crow = (const float4*)(Cm + (size_t)i * Q);
  const float4* v4 = (const float4*)v;
  for (int j = t; j < Q / 4; j += 256) {
    float4 cv = Crow[j];
    float4 vv = v4[j];
    lse_accum(m, s, (ui + vv.x - cv.x) * INV_EPS);
    lse_accum(m, s, (ui + vv.y - cv.y) * INV_EPS);
    lse_accum(m, s, (ui + vv.z - cv.z) * INV_EPS);
    lse_accum(m, s, (ui + vv.w - cv.w) * INV_EPS);
  }
  __shared__ float mr[256], sr[256];
  mr[t] = m; sr[t] = s; __syncthreads();
  for (int st = 128; st > 0; st >>= 1) {
    if (t < st) lse_merge(mr[t], sr[t], mr[t + st], sr[t + st]);
    __syncthreads();
  }
  if (t == 0) {
    float L = mr[0] + logf(sr[0]);
    u_new[i] = EPSR * (LOG_MU - L) + ui;
  }
}

// ---------- Sinkhorn: col logsumexp -> v_new (64 cols/block, float4 rows) ----------
__global__ void ot_col(const float* __restrict__ Cm, const float* __restrict__ u_new,
                       const float* __restrict__ v, float* __restrict__ v_new,
                       const int* __restrict__ done) {
  if (*done) return;
  const int t  = threadIdx.x;
  const int cg = t & 15;   // column group: 4 consecutive columns
  const int rg = t >> 4;   // row group 0..15
  const int jb = blockIdx.x * 64 + cg * 4;
  float4 vv = *(const float4*)&v[jb];
  float m0v = -INFINITY, m1v = -INFINITY, m2v = -INFINITY, m3v = -INFINITY;
  float s0 = 0.0f, s1 = 0.0f, s2 = 0.0f, s3 = 0.0f;
  for (int i = rg; i < S; i += 16) {
    float4 cv = *(const float4*)&Cm[(size_t)i * Q + jb];
    float un = u_new[i];
    lse_accum(m0v, s0, (un + vv.x - cv.x) * INV_EPS);
    lse_accum(m1v, s1, (un + vv.y - cv.y) * INV_EPS);
    lse_accum(m2v, s2, (un + vv.z - cv.z) * INV_EPS);
    lse_accum(m3v, s3, (un + vv.w - cv.w) * INV_EPS);
  }
  __shared__ float mr[256][4], sr[256][4];
  mr[t][0] = m0v; mr[t][1] = m1v; mr[t][2] = m2v; mr[t][3] = m3v;
  sr[t][0] = s0;  sr[t][1] = s1;  sr[t][2] = s2;  sr[t][3] = s3;
  __syncthreads();
  if (rg == 0) {
    float mm[4] = {m0v, m1v, m2v, m3v};
    float ss[4] = {s0, s1, s2, s3};
#pragma unroll
    for (int g = 1; g < 16; ++g) {
      int pt = g * 16 + cg;
#pragma unroll
      for (int q = 0; q < 4; ++q) lse_merge(mm[q], ss[q], mr[pt][q], sr[pt][q]);
    }
    float vj[4] = {vv.x, vv.y, vv.z, vv.w};
#pragma unroll
    for (int q = 0; q < 4; ++q) {
      float L = mm[q] + logf(ss[q]);
      v_new[jb + q] = EPSR * (LOG_NU - L) + vj[q];
    }
  }
}

// ---------- Sinkhorn: err check + commit (freeze once converged) ----------
__global__ void ot_update(float* __restrict__ u, const float* __restrict__ u_new,
                          float* __restrict__ v, const float* __restrict__ v_new,
                          int* __restrict__ done) {
  if (*done) return;
  const int t = threadIdx.x;
  float partial = 0.0f;
  for (int i = t; i < S; i += 256) partial += fabsf(u_new[i] - u[i]);
  __shared__ float red[256];
  red[t] = partial; __syncthreads();
  for (int st = 128; st > 0; st >>= 1) {
    if (t < st) red[t] += red[t + st];
    __syncthreads();
  }
  // commit updates (reference applies them before marking done)
  for (int i = t; i < S; i += 256) u[i] = u_new[i];
  for (int j = t; j < Q; j += 256) v[j] = v_new[j];
  __syncthreads();
  if (t == 0 && red[0] < THRESH) *done = 1;
}

// ---------- pi = exp(M(u,v)) in place over C; inverse row sums (float4) ----------
__global__ void ot_pi_rowsum(float* __restrict__ Cm, const float* __restrict__ u,
                             const float* __restrict__ v, float* __restrict__ inv_rs) {
  const int i = blockIdx.x;
  const int t = threadIdx.x;
  const float ui = u[i];
  float partial = 0.0f;
  float4* Crow = (float4*)(Cm + (size_t)i * Q);
  const float4* v4 = (const float4*)v;
  for (int j = t; j < Q / 4; j += 256) {
    float4 cv = Crow[j];
    float4 vv = v4[j];
    float4 p;
    p.x = expf((ui + vv.x - cv.x) * INV_EPS);
    p.y = expf((ui + vv.y - cv.y) * INV_EPS);
    p.z = expf((ui + vv.z - cv.z) * INV_EPS);
    p.w = expf((ui + vv.w - cv.w) * INV_EPS);
    Crow[j] = p;
    partial += p.x + p.y + p.z + p.w;
  }
  __shared__ float red[256];
  red[t] = partial; __syncthreads();
  for (int st = 128; st > 0; st >>= 1) {
    if (t < st) red[t] += red[t + st];
    __syncthreads();
  }
  if (t == 0) inv_rs[i] = 1.0f / red[0];
}

// ---------- GEMM2: out = diag(1/rowsum) * pi @ zq  (fp32 WMMA 16x16x4) ----------
// block tile 64(M) x 128(N), K-chunk 32 over K=4096, async double-buffered LDS
__global__ __launch_bounds__(256)
void ot_gemm_transport(const float* __restrict__ pi, const float* __restrict__ zq,
                       const float* __restrict__ inv_rs, float* __restrict__ out) {
  __shared__ float As[2][64][36];    // [buf][m][k] tile of pi
  __shared__ float Bs[2][32][132];   // [buf][k][n] tile of zq (K x N row-major)
  const int t    = threadIdx.x;
  const int wave = t >> 5;
  const int lane = t & 31;
  const int half = lane >> 4;
  const int lm   = lane & 15;
  const int mw   = wave & 3;
  const int nw0  = (wave >> 2) * 4;

  const int m0 = blockIdx.y * 64;
  const int n0 = blockIdx.x * 128;

  auto issue_chunk = [&](int k0, int buf) {
#pragma unroll
    for (int i = 0; i < 2; ++i) {
      int idx = t + 256 * i, rr = idx >> 3, k4 = idx & 7;
      async_copy16(&pi[(size_t)(m0 + rr) * Q + k0 + k4 * 4], &As[buf][rr][k4 * 4]);
    }
#pragma unroll
    for (int i = 0; i < 4; ++i) {
      int idx = t + 256 * i, kk = idx >> 5, n4 = idx & 31;
      async_copy16(&zq[(size_t)(k0 + kk) * D + n0 + n4 * 4], &Bs[buf][kk][n4 * 4]);
    }
  };

  v8f acc[4];
#pragma unroll
  for (int q = 0; q < 4; ++q) acc[q] = v8f{};

  issue_chunk(0, 0);
  wait_async0();
  __syncthreads();

  for (int k0 = 0; k0 < Q; k0 += 32) {
    const int buf = (k0 >> 5) & 1;
    const bool more = (k0 + 32) < Q;
    if (more) issue_chunk(k0 + 32, buf ^ 1);
#pragma unroll
    for (int kk = 0; kk < 32; kk += 4) {
      const int ko = kk + half * 2;
      v2f a; a.x = As[buf][mw * 16 + lm][ko]; a.y = As[buf][mw * 16 + lm][ko + 1];
#pragma unroll
      for (int q = 0; q < 4; ++q) {
        v2f b; b.x = Bs[buf][ko][(nw0 + q) * 16 + lm];
               b.y = Bs[buf][ko + 1][(nw0 + q) * 16 + lm];
        acc[q] = __builtin_amdgcn_wmma_f32_16x16x4_f32(false, a, false, b, (short)0,
                                                       acc[q], false, false);
      }
    }
    if (more) wait_async0();
    __syncthreads();
  }

#pragma unroll
  for (int r = 0; r < 8; ++r) {
    int gi = m0 + mw * 16 + r + half * 8;
    float sc = inv_rs[gi];
#pragma unroll
    for (int q = 0; q < 4; ++q) {
      out[(size_t)gi * D + n0 + (nw0 + q) * 16 + lm] = acc[q][r] * sc;
    }
  }
}

extern "C" void kernel_launch(void* const* d_in, const int* in_sizes, int n_in,
                              void* d_out, int out_size, void* d_ws, size_t ws_size,
                              hipStream_t stream) {
  const float* z_support = (const float*)d_in[0];
  const float* z_query   = (const float*)d_in[1];
  float* out = (float*)d_out;
  float* ws  = (float*)d_ws;

  float* zs     = ws + ZS_OFF;
  float* Cm     = ws + C_OFF;
  float* u      = ws + U_OFF;
  float* u_new  = ws + UN_OFF;
  float* v      = ws + V_OFF;
  float* v_new  = ws + VN_OFF;
  float* sqs    = ws + SQS_OFF;
  float* sqq    = ws + SQQ_OFF;
  float* inv_rs = ws + RS_OFF;
  int*   done   = (int*)(ws + FLAG_OFF);

  // normalized query goes straight into output slot #2
  float* zq = out + (size_t)S * D;

  ot_init<<<1, 256, 0, stream>>>(u, v, done);
  ot_normalize<<<S, 256, 0, stream>>>(z_support, zs, sqs);
  ot_normalize<<<Q, 256, 0, stream>>>(z_query, zq, sqq);

  ot_gemm_cost<<<dim3(Q / 128, S / 64), 256, 0, stream>>>(zs, zq, sqs, sqq, Cm);

  for (int it = 0; it < MAX_ITER; ++it) {
    ot_row<<<S, 256, 0, stream>>>(Cm, u, v, u_new, done);
    ot_col<<<Q / 64, 256, 0, stream>>>(Cm, u_new, v, v_new, done);
    ot_update<<<1, 256, 0, stream>>>(u, u_new, v, v_new, done);
  }

  ot_pi_rowsum<<<S, 256, 0, stream>>>(Cm, u, v, inv_rs);
  ot_gemm_transport<<<dim3(D / 128, S / 64), 256, 0, stream>>>(Cm, zq, inv_rs, out);
}